// VQVAE_46102178955961
// MI455X (gfx1250) — compile-verified
//
#include <hip/hip_runtime.h>
#include <math.h>

typedef __attribute__((ext_vector_type(2))) float f32x2;
typedef __attribute__((ext_vector_type(4))) float f32x4;
typedef __attribute__((ext_vector_type(8))) float f32x8;

#define N_TOK 65536
#define HID   512
#define EMB   256
#define KCODES 4096

// ---------------------------------------------------------------------------
// CDNA5 fp32 WMMA: D(16x16) = A(16x4) * B(4x16) + C, wave32.
// Lane l: A row = l&15, holds K = 2*(l>>4)+{0,1}; B col = l&15, same K pair.
// C/D: vgpr v -> row = v + 8*(l>>4), col = l&15.
// ---------------------------------------------------------------------------
__device__ __forceinline__ f32x8 wmma4(f32x2 a, f32x2 b, f32x8 c) {
  return __builtin_amdgcn_wmma_f32_16x16x4_f32(false, a, false, b, (short)0, c,
                                               false, false);
}

__device__ __forceinline__ float fixf(float x) {
  return __builtin_isfinite(x) ? x : 0.0f;  // nan_to_num(nan/inf -> 0)
}

// gfx1250 async global->LDS copy (ASYNCcnt tracked), per CDNA5 ISA.
__device__ __forceinline__ void async_b128(unsigned lds_off,
                                           const float* gaddr) {
  asm volatile("global_load_async_to_lds_b128 %0, %1, off" ::"v"(lds_off),
               "v"(gaddr)
               : "memory");
}
__device__ __forceinline__ void wait_async_le8() {
  asm volatile("s_wait_asynccnt 8" ::: "memory");
}
__device__ __forceinline__ void wait_async_0() {
  asm volatile("s_wait_asynccnt 0" ::: "memory");
}

// ---------------------------------------------------------------------------
// Block GEMM: C[M,N] = act(A[M,K] @ W[K,N] + bias).  K compile-time.
// Block = 128 rows (8 waves x 16) x 16 cols; B tile (K x 16) staged in LDS
// once per block via async copy (fallback scalar path for ragged/unaligned N).
// grid = (ceil(N/16), M/128)
// ---------------------------------------------------------------------------
template <int K>
__global__ __launch_bounds__(256) void gemm_bk(const float* __restrict__ A,
                                               const float* __restrict__ W,
                                               const float* __restrict__ bias,
                                               float* __restrict__ C, int N,
                                               int do_relu) {
  __shared__ float Bs[K * 16];
  int lane = threadIdx.x & 31, wave = threadIdx.x >> 5;
  int n0 = blockIdx.x << 4;
  int m0 = (blockIdx.y << 7) + (wave << 4);

  bool can_async = (n0 + 16 <= N) && ((N & 3) == 0);  // block-uniform
  if (can_async) {
    unsigned lb = (unsigned)(uintptr_t)(&Bs[0]);
    int ee = threadIdx.x * (K / 16);  // K/16 floats per thread, row-major
#pragma unroll
    for (int j = 0; j < K / 64; ++j) {
      int e2 = ee + j * 4;
      int k2 = e2 >> 4, c2 = e2 & 15;
      async_b128(lb + (unsigned)e2 * 4u, W + (size_t)k2 * N + n0 + c2);
    }
    wait_async_0();
  } else {
    for (int e = threadIdx.x; e < K * 16; e += 256) {
      int k = e >> 4, c = e & 15;
      int nc = n0 + c;
      nc = nc < N ? nc : N - 1;
      Bs[e] = W[(size_t)k * N + nc];
    }
  }
  __syncthreads();

  int row = lane & 15, kb = (lane >> 4) << 1, col = lane & 15;
  int ncol = n0 + col;
  const float* Arow = A + (size_t)(m0 + row) * K;
  f32x8 acc = {};
#pragma unroll 8
  for (int k = 0; k < K; k += 4) {
    f32x2 a = *(const f32x2*)(Arow + k + kb);
    f32x2 b;
    b.x = Bs[(k + kb) * 16 + col];
    b.y = Bs[(k + kb + 1) * 16 + col];
    acc = wmma4(a, b, acc);
  }
  int ncl = ncol < N ? ncol : N - 1;
  float bv = bias[ncl];
  if (ncol < N) {
#pragma unroll
    for (int v = 0; v < 8; ++v) {
      int r = m0 + v + ((lane >> 4) << 3);
      float val = acc[v] + bv;
      if (do_relu) val = fmaxf(val, 0.0f);
      C[(size_t)r * N + ncol] = val;
    }
  }
}

// ---------------------------------------------------------------------------
// Fold affine pre-relu chain:
//   Wc[64x512]  = cont_proj_w @ fuse_w1[512:1024]
//   Wk[12x512]  = cat_proj_w  @ fuse_w1[1024:1536]
//   bfused[512] = fuse_b1 + cont_proj_b@mid + cat_proj_b@hi
// ---------------------------------------------------------------------------
__global__ void fold_kernel(const float* __restrict__ cw,
                            const float* __restrict__ cbv,
                            const float* __restrict__ kw,
                            const float* __restrict__ kbv,
                            const float* __restrict__ fuse_w1,
                            const float* __restrict__ fuse_b1,
                            float* __restrict__ Wc, float* __restrict__ Wk,
                            float* __restrict__ bfused) {
  int i = blockIdx.x * blockDim.x + threadIdx.x;
  if (i >= 77 * HID) return;
  int r = i >> 9, c = i & (HID - 1);
  const float* fmid = fuse_w1 + (size_t)HID * HID;
  const float* fhi = fuse_w1 + (size_t)2 * HID * HID;
  if (r < 64) {
    float s = 0.f;
    for (int k = 0; k < HID; ++k) s += cw[r * HID + k] * fmid[k * HID + c];
    Wc[r * HID + c] = s;
  } else if (r < 76) {
    int rr = r - 64;
    float s = 0.f;
    for (int k = 0; k < HID; ++k) s += kw[rr * HID + k] * fhi[k * HID + c];
    Wk[rr * HID + c] = s;
  } else {
    float s = fuse_b1[c];
    for (int k = 0; k < HID; ++k)
      s += cbv[k] * fmid[k * HID + c] + kbv[k] * fhi[k * HID + c];
    bfused[c] = s;
  }
}

// naip conv (3x3, 1 band) + relu -> (256, 512)
__global__ void naip_conv_kernel(const float* __restrict__ naip,
                                 const float* __restrict__ conv_w,
                                 const float* __restrict__ conv_b,
                                 float* __restrict__ feat) {
  int i = blockIdx.x * blockDim.x + threadIdx.x;
  if (i >= 256 * HID) return;
  int b = i >> 9, h = i & (HID - 1);
  float s = conv_b[h];
  for (int j = 0; j < 9; ++j) s += fixf(naip[b * 9 + j]) * conv_w[h * 9 + j];
  feat[i] = fmaxf(s, 0.0f);
}

// ---------------------------------------------------------------------------
// Fused encoder: per block = 16 tokens.
// Stage 1: h1 = relu(cont@Wc + catE@Wk + naip_base[b])  -> LDS (never to HBM)
// Stage 2: z_e = h1 @ fuse_w2 + fuse_b2                 -> global
// ---------------------------------------------------------------------------
__global__ __launch_bounds__(256) void encoder_kernel(
    const float* __restrict__ cont,       // 65536 x 64
    const int* __restrict__ cat,          // 65536 x 2
    const float* __restrict__ Wc,         // 64 x 512
    const float* __restrict__ Wk,         // 12 x 512
    const float* __restrict__ naip_base,  // 256 x 512 (incl. bfused)
    const float* __restrict__ spec_emb, const float* __restrict__ soil_emb,
    const float* __restrict__ fuse_w2,  // 512 x 256
    const float* __restrict__ fuse_b2,  // 256
    float* __restrict__ z_e)            // 65536 x 256
{
  __shared__ float h1s[16 * 516];  // stride 516 staggers banks
  __shared__ float catE[16 * 12];
  int m0 = blockIdx.x << 4;
  int bb = m0 >> 8;  // batch index (uniform per block)
  int lane = threadIdx.x & 31;
  int wave = threadIdx.x >> 5;

  if (threadIdx.x < 192) {  // stage cat embeddings for 16 rows
    int r = threadIdx.x / 12, k = threadIdx.x % 12;
    int t = m0 + r;
    catE[r * 12 + k] = (k < 6) ? spec_emb[cat[2 * t] * 6 + k]
                               : soil_emb[cat[2 * t + 1] * 6 + (k - 6)];
  }
  __syncthreads();

  int row = lane & 15;
  int kb = (lane >> 4) << 1;
  const float* Arow = cont + (size_t)(m0 + row) * 64;

  // ---- stage 1: each wave does 4 n-tiles covering N=512
  for (int j = 0; j < 4; ++j) {
    int n0 = wave * 64 + j * 16;
    int col = n0 + (lane & 15);
    f32x8 acc = {};
#pragma unroll
    for (int k = 0; k < 64; k += 4) {
      f32x2 a = *(const f32x2*)(Arow + k + kb);
      a.x = fixf(a.x);
      a.y = fixf(a.y);
      f32x2 b;
      b.x = Wc[(k + kb) * HID + col];
      b.y = Wc[(k + kb + 1) * HID + col];
      acc = wmma4(a, b, acc);
    }
    float nb = naip_base[bb * HID + col];
#pragma unroll
    for (int v = 0; v < 8; ++v) {
      int r = v + ((lane >> 4) << 3);
      float s = acc[v] + nb;
#pragma unroll
      for (int kk = 0; kk < 12; ++kk)
        s += catE[r * 12 + kk] * Wk[kk * HID + col];
      h1s[r * 516 + col] = fmaxf(s, 0.0f);
    }
  }
  __syncthreads();

  // ---- stage 2: each wave does 2 n-tiles covering N=256, K=512 from LDS
  for (int j = 0; j < 2; ++j) {
    int n0 = wave * 32 + j * 16;
    int col = n0 + (lane & 15);
    f32x8 acc = {};
#pragma unroll 8
    for (int k = 0; k < HID; k += 4) {
      f32x2 a, b;
      a.x = h1s[row * 516 + k + kb];
      a.y = h1s[row * 516 + k + kb + 1];
      b.x = fuse_w2[(k + kb) * EMB + col];
      b.y = fuse_w2[(k + kb + 1) * EMB + col];
      acc = wmma4(a, b, acc);
    }
    float bv = fuse_b2[col];
#pragma unroll
    for (int v = 0; v < 8; ++v) {
      int r = m0 + v + ((lane >> 4) << 3);
      z_e[(size_t)r * EMB + col] = acc[v] + bv;
    }
  }
}

// row squared-norms (32 lanes per row)
__global__ void rownorm_kernel(const float* __restrict__ X,
                               float* __restrict__ out, int rows, int cols) {
  int lane = threadIdx.x & 31;
  int r = (blockIdx.x * blockDim.x + threadIdx.x) >> 5;
  if (r >= rows) return;
  const float* xr = X + (size_t)r * cols;
  float s = 0.f;
  for (int k = lane; k < cols; k += 32) {
    float v = xr[k];
    s += v * v;
  }
#pragma unroll
  for (int off = 16; off; off >>= 1) s += __shfl_xor(s, off, 32);
  if (lane == 0) out[r] = s;
}

__global__ void zero_kernel(float* __restrict__ p, int n) {
  int i = blockIdx.x * blockDim.x + threadIdx.x;
  if (i < n) p[i] = 0.0f;
}

// ---------------------------------------------------------------------------
// VQ argmin: scores = cnorm[c] - 2 * (z_e @ cb^T), streamed as WMMA GEMM.
// Block = 128 tokens (8 waves x 16). Codebook chunks of 32 codes are
// double-buffered in LDS via global_load_async_to_lds_b128 (ASYNCcnt),
// one A fragment feeds two WMMA tiles per k-step.
// ---------------------------------------------------------------------------
__device__ __forceinline__ void vq_stage(unsigned lds_base,
                                         const float* __restrict__ cb, int c0,
                                         int tid) {
  int rr = tid >> 3;             // code row in chunk: 0..31
  int cc = (tid & 7) * 32;       // start k: 32 floats per thread
  const float* g = cb + (size_t)(c0 + rr) * EMB + cc;
  unsigned l = lds_base + (unsigned)(rr * 260 + cc) * 4u;
#pragma unroll
  for (int j = 0; j < 8; ++j) async_b128(l + j * 16u, g + j * 4);
}

__global__ __launch_bounds__(256) void vq_kernel(
    const float* __restrict__ z_e, const float* __restrict__ cb,
    const float* __restrict__ cnorm, const float* __restrict__ znorm,
    int* __restrict__ idx, float* __restrict__ counts,
    float* __restrict__ vqsum) {
  __shared__ float cbs[2][32 * 260];  // padded rows: conflict-free b64 reads
  unsigned lb0 = (unsigned)(uintptr_t)(&cbs[0][0]);
  unsigned lb1 = (unsigned)(uintptr_t)(&cbs[1][0]);

  int lane = threadIdx.x & 31;
  int wave = threadIdx.x >> 5;
  int m0 = (blockIdx.x * 8 + wave) << 4;
  int row = lane & 15;
  int kb = (lane >> 4) << 1;
  int col = lane & 15;
  const float* Arow = z_e + (size_t)(m0 + row) * EMB;

  float best[8];
  int bidx[8];
#pragma unroll
  for (int v = 0; v < 8; ++v) {
    best[v] = 3.4e38f;
    bidx[v] = 0;
  }

  vq_stage(lb0, cb, 0, threadIdx.x);
  const int NCH = KCODES / 32;  // 128 chunks
  for (int ci = 0; ci < NCH; ++ci) {
    int cur = ci & 1;
    if (ci + 1 < NCH) {
      vq_stage(cur ? lb0 : lb1, cb, (ci + 1) * 32, threadIdx.x);
      wait_async_le8();  // chunk ci complete (in-order done)
    } else {
      wait_async_0();
    }
    __syncthreads();

    const float* B0 = &cbs[cur][col * 260];
    const float* B1 = &cbs[cur][(col + 16) * 260];
    f32x8 acc0 = {}, acc1 = {};
#pragma unroll 8
    for (int k = 0; k < EMB; k += 4) {
      f32x2 a = *(const f32x2*)(Arow + k + kb);
      f32x2 b0 = *(const f32x2*)(B0 + k + kb);
      f32x2 b1 = *(const f32x2*)(B1 + k + kb);
      acc0 = wmma4(a, b0, acc0);
      acc1 = wmma4(a, b1, acc1);
    }
    int c0 = ci * 32;
    float cn0 = cnorm[c0 + col];
    float cn1 = cnorm[c0 + 16 + col];
#pragma unroll
    for (int v = 0; v < 8; ++v) {
      float s0 = cn0 - 2.0f * acc0[v];
      if (s0 < best[v]) {
        best[v] = s0;
        bidx[v] = c0 + col;
      }
      float s1 = cn1 - 2.0f * acc1[v];
      if (s1 < best[v]) {
        best[v] = s1;
        bidx[v] = c0 + 16 + col;
      }
    }
    __syncthreads();
  }

  // reduce min over the 16 lanes sharing each row (xor<=8 stays in half-wave)
  float lsum = 0.0f;
#pragma unroll
  for (int v = 0; v < 8; ++v) {
    float bv = best[v];
    int bi = bidx[v];
#pragma unroll
    for (int off = 8; off; off >>= 1) {
      float ov = __shfl_xor(bv, off, 32);
      int oi = __shfl_xor(bi, off, 32);
      if (ov < bv || (ov == bv && oi < bi)) {
        bv = ov;
        bi = oi;
      }
    }
    if ((lane & 15) == 0) {
      int t = m0 + v + ((lane >> 4) << 3);
      idx[t] = bi;
      atomicAdd(&counts[bi], 1.0f);
      lsum += znorm[t] + bv;  // ||z_e - cb[bi]||^2
    }
  }
  if ((lane & 15) == 0) atomicAdd(vqsum, lsum);
}

// gather decoder tables -> outputs (pure streaming write, tables L2-hot)
__global__ __launch_bounds__(256) void output_kernel(
    const int* __restrict__ idx, const float* __restrict__ cont_tab,
    const float* __restrict__ spec_tab, const float* __restrict__ soil_tab,
    float* __restrict__ out) {
  int t = blockIdx.x;
  int code = idx[t];
  __builtin_prefetch(spec_tab + (size_t)code * 1000, 0, 0);
  float* cont_pred = out + (size_t)t * 64;
  float* spec = out + (size_t)N_TOK * 64 + (size_t)t * 1000;
  float* soil = out + (size_t)N_TOK * 64 + (size_t)N_TOK * 1000 +
                (size_t)t * 50;
  int j = threadIdx.x;
  if (j < 16)
    ((f32x4*)cont_pred)[j] = ((const f32x4*)(cont_tab + code * 64))[j];
  if (j < 250)
    ((f32x4*)spec)[j] = ((const f32x4*)(spec_tab + (size_t)code * 1000))[j];
  if (j < 25)
    ((f32x2*)soil)[j] = ((const f32x2*)(soil_tab + code * 50))[j];
}

__global__ __launch_bounds__(256) void canopy_kernel(
    const int* __restrict__ idx, const float* __restrict__ h2_tab,
    const float* __restrict__ w1, const float* __restrict__ b1,
    const float* __restrict__ w2, const float* __restrict__ b2,
    float* __restrict__ canopy) {
  __shared__ float cin[HID];
  __shared__ float hc[HID];
  __shared__ float red[8];
  int bb = blockIdx.x;
  int code = idx[bb * 256 + 255];  // h[:, -1, :]
  for (int j = threadIdx.x; j < HID; j += 256)
    cin[j] = h2_tab[(size_t)code * HID + j];
  __syncthreads();
  for (int j = threadIdx.x; j < HID; j += 256) {
    float s = b1[j];
    for (int k = 0; k < HID; ++k) s += cin[k] * w1[k * HID + j];
    hc[j] = fmaxf(s, 0.0f);
  }
  __syncthreads();
  float s = 0.f;
  for (int j = threadIdx.x; j < HID; j += 256) s += hc[j] * w2[j];
#pragma unroll
  for (int off = 16; off; off >>= 1) s += __shfl_xor(s, off, 32);
  int lane = threadIdx.x & 31, wave = threadIdx.x >> 5;
  if (lane == 0) red[wave] = s;
  __syncthreads();
  if (threadIdx.x == 0) {
    float tot = b2[0];
    for (int w = 0; w < 8; ++w) tot += red[w];
    canopy[bb] = tot;
  }
}

__global__ void finalize_kernel(const float* __restrict__ counts,
                                const float* __restrict__ vqsum,
                                float* __restrict__ out_vq,
                                float* __restrict__ out_ppl) {
  __shared__ float red[8];
  float s = 0.f;
  for (int i = threadIdx.x; i < KCODES; i += 256) {
    float p = counts[i] * (1.0f / (float)N_TOK);
    s += p * logf(p + 1e-12f);
  }
#pragma unroll
  for (int off = 16; off; off >>= 1) s += __shfl_xor(s, off, 32);
  int lane = threadIdx.x & 31, wave = threadIdx.x >> 5;
  if (lane == 0) red[wave] = s;
  __syncthreads();
  if (threadIdx.x == 0) {
    float H = 0.f;
    for (int w = 0; w < 8; ++w) H += red[w];
    *out_ppl = expf(-H);
    *out_vq = 1.25f * vqsum[0] / ((float)N_TOK * (float)EMB);
  }
}

// ---------------------------------------------------------------------------
extern "C" void kernel_launch(void* const* d_in, const int* in_sizes, int n_in,
                              void* d_out, int out_size, void* d_ws,
                              size_t ws_size, hipStream_t stream) {
  (void)in_sizes;
  (void)n_in;
  (void)out_size;
  (void)ws_size;
  const float* cont = (const float*)d_in[0];
  const int* cat = (const int*)d_in[1];
  const float* naip = (const float*)d_in[2];
  const float* conv_w = (const float*)d_in[3];
  const float* conv_b = (const float*)d_in[4];
  const float* naip_proj_w = (const float*)d_in[5];
  const float* naip_proj_b = (const float*)d_in[6];
  const float* cont_proj_w = (const float*)d_in[7];
  const float* cont_proj_b = (const float*)d_in[8];
  const float* species_emb = (const float*)d_in[9];
  const float* soil_emb = (const float*)d_in[10];
  const float* cat_proj_w = (const float*)d_in[11];
  const float* cat_proj_b = (const float*)d_in[12];
  const float* fuse_w1 = (const float*)d_in[13];
  const float* fuse_b1 = (const float*)d_in[14];
  const float* fuse_w2 = (const float*)d_in[15];
  const float* fuse_b2 = (const float*)d_in[16];
  const float* codebook = (const float*)d_in[17];
  const float* dec_w1 = (const float*)d_in[18];
  const float* dec_b1 = (const float*)d_in[19];
  const float* dec_w2 = (const float*)d_in[20];
  const float* dec_b2 = (const float*)d_in[21];
  const float* cont_head_w = (const float*)d_in[22];
  const float* cont_head_b = (const float*)d_in[23];
  const float* spec_head_w = (const float*)d_in[24];
  const float* spec_head_b = (const float*)d_in[25];
  const float* soil_head_w = (const float*)d_in[26];
  const float* soil_head_b = (const float*)d_in[27];
  const float* canopy_w1 = (const float*)d_in[28];
  const float* canopy_b1 = (const float*)d_in[29];
  const float* canopy_w2 = (const float*)d_in[30];
  const float* canopy_b2 = (const float*)d_in[31];
  float* out = (float*)d_out;

  // workspace layout (fp32 elements), ~100 MB total
  float* ws = (float*)d_ws;
  size_t o = 0;
  float* Wc = ws + o;
  o += 64 * HID;
  float* Wk = ws + o;
  o += 12 * HID;
  float* bfused = ws + o;
  o += HID;
  float* naip_feat = ws + o;
  o += 256 * HID;
  float* naip_feat2 = ws + o;
  o += 256 * HID;
  float* naip_base = ws + o;
  o += 256 * HID;
  float* z_e = ws + o;
  o += (size_t)N_TOK * EMB;
  float* znorm = ws + o;
  o += N_TOK;
  float* cnorm = ws + o;
  o += KCODES;
  float* counts = ws + o;
  o += KCODES;
  float* vqsum = ws + o;
  o += 1;  // contiguous with counts (zeroed together)
  int* idx = (int*)(ws + o);
  o += N_TOK;
  float* h1_tab = ws + o;
  o += (size_t)KCODES * HID;
  float* h2_tab = ws + o;
  o += (size_t)KCODES * HID;
  float* cont_tab = ws + o;
  o += (size_t)KCODES * 64;
  float* spec_tab = ws + o;
  o += (size_t)KCODES * 1000;
  float* soil_tab = ws + o;
  o += (size_t)KCODES * 50;

  auto gemm = [&](const float* A, const float* W, const float* bias, float* C,
                  int M, int N, int K, int relu) {
    dim3 g((N + 15) / 16, M / 128);
    if (K == 256)
      gemm_bk<256><<<g, 256, 0, stream>>>(A, W, bias, C, N, relu);
    else
      gemm_bk<512><<<g, 256, 0, stream>>>(A, W, bias, C, N, relu);
  };

  // encoder prep
  fold_kernel<<<(77 * HID + 255) / 256, 256, 0, stream>>>(
      cont_proj_w, cont_proj_b, cat_proj_w, cat_proj_b, fuse_w1, fuse_b1, Wc,
      Wk, bfused);
  naip_conv_kernel<<<(256 * HID + 255) / 256, 256, 0, stream>>>(
      naip, conv_w, conv_b, naip_feat);
  gemm(naip_feat, naip_proj_w, naip_proj_b, naip_feat2, 256, HID, HID, 0);
  gemm(naip_feat2, fuse_w1 /*rows 0..511*/, bfused, naip_base, 256, HID, HID,
       0);

  // fused encoder -> z_e
  encoder_kernel<<<N_TOK / 16, 256, 0, stream>>>(cont, cat, Wc, Wk, naip_base,
                                                 species_emb, soil_emb,
                                                 fuse_w2, fuse_b2, z_e);

  // VQ
  rownorm_kernel<<<N_TOK / 8, 256, 0, stream>>>(z_e, znorm, N_TOK, EMB);
  rownorm_kernel<<<KCODES / 8, 256, 0, stream>>>(codebook, cnorm, KCODES, EMB);
  zero_kernel<<<(KCODES + 1 + 255) / 256, 256, 0, stream>>>(counts,
                                                            KCODES + 1);
  vq_kernel<<<N_TOK / 128, 256, 0, stream>>>(z_e, codebook, cnorm, znorm, idx,
                                             counts, vqsum);

  // decoder tables on the 4096 codebook rows only
  gemm(codebook, dec_w1, dec_b1, h1_tab, KCODES, HID, EMB, 1);
  gemm(h1_tab, dec_w2, dec_b2, h2_tab, KCODES, HID, HID, 1);
  gemm(h2_tab, cont_head_w, cont_head_b, cont_tab, KCODES, 64, HID, 0);
  gemm(h2_tab, spec_head_w, spec_head_b, spec_tab, KCODES, 1000, HID, 0);
  gemm(h2_tab, soil_head_w, soil_head_b, soil_tab, KCODES, 50, HID, 0);

  // outputs
  output_kernel<<<N_TOK, 256, 0, stream>>>(idx, cont_tab, spec_tab, soil_tab,
                                           out);
  const size_t OFF_CAN =
      (size_t)N_TOK * 64 + (size_t)N_TOK * 1000 + (size_t)N_TOK * 50;
  canopy_kernel<<<256, 256, 0, stream>>>(idx, h2_tab, canopy_w1, canopy_b1,
                                         canopy_w2, canopy_b2, out + OFF_CAN);
  finalize_kernel<<<1, 256, 0, stream>>>(counts, vqsum, out + OFF_CAN + 256,
                                         out + OFF_CAN + 257);
}